// YOLOv1Loss2D_53360673686126
// MI455X (gfx1250) — compile-verified
//
#include <hip/hip_runtime.h>
#include <hip/hip_bf16.h>
#include <stdint.h>

// ---------------------------------------------------------------------------
// YOLOv1 loss on MI455X (gfx1250).
// HBM-bound streaming reduction: ~193 MB read, ~8.3 us floor at 23.3 TB/s.
// Strategy: async global->LDS DMA (ASYNCcnt) double-buffered tiles of 128
// cells; per-thread cell loss from LDS; WMMA f32 16x16x4 ones-matmul wave
// reduction; deterministic two-pass final sum.
// ---------------------------------------------------------------------------

#define CELL_F   30
#define TILE     128                     // cells per tile
#define BLOCK    128                     // threads per block (4 waves)
#define TILE_B   (TILE * CELL_F * 4)     // 15360 bytes per array per tile
#define XFERS    (TILE_B / 16)           // 960 16-byte transfers per array
#define ROUNDS   (XFERS / 64)            // 15: 64 threads per array stream

typedef __attribute__((ext_vector_type(2))) float v2f;
typedef __attribute__((ext_vector_type(8))) float v8f;

// Low 32 bits of a flat shared-aperture address == LDS byte offset.
__device__ __forceinline__ uint32_t lds_off(const void* p) {
    return (uint32_t)(uintptr_t)p;
}

// CDNA5 async DMA: 16 bytes per lane, global (SADDR base + VGPR offset) -> LDS.
__device__ __forceinline__ void async_cp16(uint32_t ldsAddr, const void* gbase,
                                           uint32_t gOff) {
    asm volatile("global_load_async_to_lds_b128 %0, %1, %2"
                 :: "v"(ldsAddr), "v"(gOff), "s"(gbase)
                 : "memory");
}

__device__ __forceinline__ void wait_async_15() {
    asm volatile("s_wait_asynccnt 0xf" ::: "memory");
}
__device__ __forceinline__ void wait_async_0() {
    asm volatile("s_wait_asynccnt 0x0" ::: "memory");
}

// Stage one tile (y_true slice + y_pred slice) into LDS.
// Waves 0-1 stream y_true, waves 2-3 stream y_pred: per-wave uniform SADDR,
// exactly 15 async issues per lane per tile.
__device__ __forceinline__ void stage(const float* gT, const float* gP,
                                      float* sT, float* sP,
                                      uint32_t tileByte, uint32_t maxByte,
                                      int tid) {
    const uint32_t lane16 = (uint32_t)(tid & 63) * 16u;
    if (tid < 64) {
        const uint32_t lb = lds_off(sT);
#pragma unroll
        for (int r = 0; r < ROUNDS; ++r) {
            uint32_t off = (uint32_t)(r * 64) * 16u + lane16;
            uint32_t g = tileByte + off;
            if (g > maxByte) g = maxByte;
            async_cp16(lb + off, gT, g);
        }
    } else {
        const uint32_t lb = lds_off(sP);
#pragma unroll
        for (int r = 0; r < ROUNDS; ++r) {
            uint32_t off = (uint32_t)(r * 64) * 16u + lane16;
            uint32_t g = tileByte + off;
            if (g > maxByte) g = maxByte;
            async_cp16(lb + off, gP, g);
        }
    }
}

// Wave32 sum via V_WMMA_F32_16X16X4_F32 with B == ones:
// A[m][0]=acc(lane m), A[m][2]=acc(lane m+16) -> D rows replicate row sums.
// Sum of 8 D VGPRs at lane L plus lane L^16 gives the full wave sum.
__device__ __forceinline__ float wave_reduce(float acc) {
#if __has_builtin(__builtin_amdgcn_wmma_f32_16x16x4_f32)
    v2f a; a[0] = acc;  a[1] = 0.0f;
    v2f b; b[0] = 1.0f; b[1] = 1.0f;
    v8f c = {};
    c = __builtin_amdgcn_wmma_f32_16x16x4_f32(
            /*neg_a=*/false, a, /*neg_b=*/false, b,
            /*c_mod=*/(short)0, c, /*reuse_a=*/false, /*reuse_b=*/false);
    float s = c[0] + c[1] + c[2] + c[3] + c[4] + c[5] + c[6] + c[7];
    s += __shfl_xor(s, 16, 32);
    return s;
#else
    for (int off = 16; off > 0; off >>= 1) acc += __shfl_xor(acc, off, 32);
    return acc;
#endif
}

__device__ __forceinline__ float iou_yolo(float cx1, float cy1, float w1, float h1,
                                          float cx2, float cy2, float w2, float h2) {
    float ax1 = cx1 - w1 * 0.5f, ay1 = cy1 - h1 * 0.5f;
    float ax2 = cx1 + w1 * 0.5f, ay2 = cy1 + h1 * 0.5f;
    float bx1 = cx2 - w2 * 0.5f, by1 = cy2 - h2 * 0.5f;
    float bx2 = cx2 + w2 * 0.5f, by2 = cy2 + h2 * 0.5f;
    float iw = fmaxf(fminf(ax2, bx2) - fmaxf(ax1, bx1), 0.0f);
    float ih = fmaxf(fminf(ay2, by2) - fmaxf(ay1, by1), 0.0f);
    float inter = iw * ih;
    float a1 = fabsf((ax2 - ax1) * (ay2 - ay1));
    float a2 = fabsf((bx2 - bx1) * (by2 - by1));
    return inter / (a1 + a2 - inter + 1e-6f);
}

__device__ __forceinline__ float cell_loss(const float* t, const float* p) {
    float iou1 = iou_yolo(t[0], t[1], t[2], t[3], p[0], p[1], p[2], p[3]);
    float iou2 = iou_yolo(t[0], t[1], t[2], t[3], p[5], p[6], p[7], p[8]);
    bool best1 = iou1 > iou2;
    float bx = best1 ? p[0] : p[5];
    float by = best1 ? p[1] : p[6];
    float bw = best1 ? p[2] : p[7];
    float bh = best1 ? p[3] : p[8];
    float ch = best1 ? p[4] : p[9];   // responsible conf
    float oh = best1 ? p[9] : p[4];   // other conf

    float dx = t[0] - bx, dy = t[1] - by;
    float xy = dx * dx + dy * dy;
    float sw = sqrtf(t[2]) - sqrtf(fabsf(bw + 1e-6f));
    float sh = sqrtf(t[3]) - sqrtf(fabsf(bh + 1e-6f));
    float wh = sw * sw + sh * sh;
    float oc = t[4] - ch; oc *= oc;
    float noobj_in = 0.5f * oh * oh;
    float d4 = t[4] - p[4], d9 = t[4] - p[9];
    float noobj_cells = 0.5f * (d4 * d4 + d9 * d9);
    float cls = 0.0f;
#pragma unroll
    for (int c = 10; c < 30; ++c) {
        float d = t[c] - p[c];
        cls += d * d;
    }
    float obj = (t[4] == 1.0f) ? 1.0f : 0.0f;
    return obj * (5.0f * (xy + wh) + oc + noobj_in + cls)
         + (1.0f - obj) * noobj_cells;
}

__global__ __launch_bounds__(BLOCK)
void yolo_main(const float* __restrict__ yt, const float* __restrict__ yp,
               float* __restrict__ partial, int ncells, int ntiles) {
    __shared__ __align__(16) float sm[2][2][TILE * CELL_F]; // 61440 B double buf
    __shared__ float wsum[BLOCK / 32];

    const int tid = threadIdx.x;
    const uint32_t maxByte = ((uint32_t)ncells * (CELL_F * 4u) - 16u) & ~15u;
    float acc = 0.0f;

    const int t0 = blockIdx.x;
    const int stride = gridDim.x;

    if (t0 < ntiles)
        stage(yt, yp, sm[0][0], sm[0][1], (uint32_t)t0 * TILE_B, maxByte, tid);

    int buf = 0;
    for (int t = t0; t < ntiles; t += stride) {
        const int tn = t + stride;
        if (tn < ntiles) {  // prefetch next tile into other buffer
            stage(yt, yp, sm[buf ^ 1][0], sm[buf ^ 1][1],
                  (uint32_t)tn * TILE_B, maxByte, tid);
            wait_async_15();   // tile t's 15 issues done; tn's may be in flight
        } else {
            wait_async_0();
        }
        __syncthreads();       // all waves' DMA for tile t landed

        const int cell = t * TILE + tid;
        if (cell < ncells) {
            const float* ct = &sm[buf][0][tid * CELL_F];
            const float* cp = &sm[buf][1][tid * CELL_F];
            acc += cell_loss(ct, cp);
        }
        __syncthreads();       // done reading buf before it is overwritten
        buf ^= 1;
    }

    float w = wave_reduce(acc);
    if ((tid & 31) == 0) wsum[tid >> 5] = w;
    __syncthreads();
    if (tid == 0) {
        float tot = 0.0f;
#pragma unroll
        for (int i = 0; i < BLOCK / 32; ++i) tot += wsum[i];
        partial[blockIdx.x] = tot;
    }
}

__global__ __launch_bounds__(256)
void yolo_finalize(const float* __restrict__ part, int n,
                   float* __restrict__ out, float scale) {
    __shared__ float wsum[8];
    float acc = 0.0f;
    for (int i = threadIdx.x; i < n; i += 256) acc += part[i];
    float w = wave_reduce(acc);
    if ((threadIdx.x & 31) == 0) wsum[threadIdx.x >> 5] = w;
    __syncthreads();
    if (threadIdx.x == 0) {
        float tot = 0.0f;
#pragma unroll
        for (int i = 0; i < 8; ++i) tot += wsum[i];
        out[0] = tot * scale;
    }
}

extern "C" void kernel_launch(void* const* d_in, const int* in_sizes, int n_in,
                              void* d_out, int out_size, void* d_ws, size_t ws_size,
                              hipStream_t stream) {
    const float* yt = (const float*)d_in[0];
    const float* yp = (const float*)d_in[1];
    float* out = (float*)d_out;
    float* partial = (float*)d_ws;

    const int total  = in_sizes[0];          // batch*7*7*30
    const int ncells = total / CELL_F;       // 802816
    const int batch  = ncells / 49;          // 16384
    const int ntiles = (ncells + TILE - 1) / TILE;
    const int blocks = ntiles < 1024 ? ntiles : 1024;

    yolo_main<<<blocks, BLOCK, 0, stream>>>(yt, yp, partial, ncells, ntiles);
    yolo_finalize<<<1, 256, 0, stream>>>(partial, blocks, out, 1.0f / (float)batch);
}